// TransformerEmbedDecoder_78520592106179
// MI455X (gfx1250) — compile-verified
//
#include <hip/hip_runtime.h>
#include <hip/hip_bf16.h>
#include <math.h>

// Sizes from the reference
#define NIMG 2
#define BATCH 32
#define DIM 512
#define HWPX 324          // 18*18
#define LQ 648            // NIMG*HWPX (== L_mem)
#define KD 128
#define LPAD 672          // 21*32, padded score stride for 2x2 wave tiles
#define MROWS 20736       // LQ*BATCH
#define NELEM 10616832    // LQ*BATCH*DIM
#define NORM_SCALE 0.0110485434560398f   // sqrt(1/(512*16))
#define INORM_N 165888.0f                // DIM*H*W
#define TEMP 30.0f

typedef float v2f __attribute__((ext_vector_type(2)));
typedef float v8f __attribute__((ext_vector_type(8)));

static __device__ __forceinline__ v8f wmma4(v2f a, v2f b, v8f c) {
  // D = A(16x4,f32) * B(4x16,f32) + C(16x16,f32); wave32 CDNA5 WMMA
  return __builtin_amdgcn_wmma_f32_16x16x4_f32(false, a, false, b, (short)0, c, false, false);
}

// ---------------- setup / elementwise kernels ----------------

// dst[(l*32+b)*512+d] = src[((i*32+b)*512+d)*324+hw],  l = i*324+hw
__global__ void k_flatten(const float* __restrict__ src, float* __restrict__ dst) {
  size_t o = (size_t)blockIdx.x * blockDim.x + threadIdx.x;
  if (o >= NELEM) return;
  int d = (int)(o & 511);
  int b = (int)((o >> 9) & 31);
  int l = (int)(o >> 14);
  int i = l / HWPX, hw = l - i * HWPX;
  dst[o] = src[(((size_t)(i * 32 + b) * 512 + d) * HWPX) + hw];
}

// x += 0.001 * pos_embed (pos_embed read in raw layout)
__global__ void k_addpe(float* __restrict__ x, const float* __restrict__ pe) {
  size_t o = (size_t)blockIdx.x * blockDim.x + threadIdx.x;
  if (o >= NELEM) return;
  int d = (int)(o & 511);
  int b = (int)((o >> 9) & 31);
  int l = (int)(o >> 14);
  int i = l / HWPX, hw = l - i * HWPX;
  x[o] += 0.001f * pe[(((size_t)(i * 32 + b) * 512 + d) * HWPX) + hw];
}

// p[(i*324+hw)*32+b] = pos[(i*32+b)*324+hw]
__global__ void k_pos(const float* __restrict__ pos, float* __restrict__ p) {
  int t = blockIdx.x * blockDim.x + threadIdx.x;
  if (t >= LQ * BATCH) return;
  int b = t & 31, l = t >> 5;
  int i = l / HWPX, hw = l - i * HWPX;
  p[t] = pos[(size_t)(i * 32 + b) * HWPX + hw];
}

__global__ void k_memp(const float* __restrict__ mem, const float* __restrict__ p,
                       float* __restrict__ memp) {
  size_t o = (size_t)blockIdx.x * blockDim.x + threadIdx.x;
  if (o >= NELEM) return;
  memp[o] = mem[o] * p[o >> 9];
}

// y = x + y  (in place into y)
__global__ void k_addinto(float* __restrict__ y, const float* __restrict__ x) {
  size_t o = (size_t)blockIdx.x * blockDim.x + threadIdx.x;
  if (o >= NELEM) return;
  y[o] = x[o] + y[o];
}

// ------ projection GEMM (2x2 register-blocked): raw[M,128] = X[M,512]*W[128,512]^T + bias ------
__global__ void k_proj(const float* __restrict__ X, const float* __restrict__ W,
                       const float* __restrict__ bias, float* __restrict__ raw) {
  int lane = threadIdx.x;
  int half = lane >> 4, l15 = lane & 15;
  int m0 = blockIdx.x * 32, n0 = blockIdx.y * 32;
  const float* a0 = X + (size_t)(m0 + l15) * 512 + 2 * half;
  const float* a1 = X + (size_t)(m0 + 16 + l15) * 512 + 2 * half;
  const float* b0 = W + (size_t)(n0 + l15) * 512 + 2 * half;
  const float* b1 = W + (size_t)(n0 + 16 + l15) * 512 + 2 * half;
  v8f c00 = {}, c01 = {}, c10 = {}, c11 = {};
  for (int k0 = 0; k0 < 512; k0 += 4) {
    v2f A0 = {a0[k0], a0[k0 + 1]};
    v2f A1 = {a1[k0], a1[k0 + 1]};
    v2f B0 = {b0[k0], b0[k0 + 1]};
    v2f B1 = {b1[k0], b1[k0 + 1]};
    c00 = wmma4(A0, B0, c00);
    c01 = wmma4(A0, B1, c01);
    c10 = wmma4(A1, B0, c10);
    c11 = wmma4(A1, B1, c11);
  }
  float bn0 = bias[n0 + l15], bn1 = bias[n0 + 16 + l15];
  int mrow = m0 + 8 * half;
  for (int r = 0; r < 8; ++r) {
    raw[(size_t)(mrow + r) * KD + n0 + l15]           = c00[r] + bn0;
    raw[(size_t)(mrow + r) * KD + n0 + 16 + l15]      = c01[r] + bn1;
    raw[(size_t)(mrow + 16 + r) * KD + n0 + l15]      = c10[r] + bn0;
    raw[(size_t)(mrow + 16 + r) * KD + n0 + 16 + l15] = c11[r] + bn1;
  }
}

// l2-normalize each 128-wide row; relayout [l*32+b][k] -> [b][l][k]
__global__ void k_l2row(const float* __restrict__ raw, float* __restrict__ out) {
  int wave = threadIdx.x >> 5, lane = threadIdx.x & 31;
  int m = blockIdx.x * 4 + wave;           // 20736 rows total
  float4 v = ((const float4*)(raw + (size_t)m * KD))[lane];
  float ss = v.x * v.x + v.y * v.y + v.z * v.z + v.w * v.w;
  for (int o = 16; o > 0; o >>= 1) ss += __shfl_xor(ss, o, 32);
  float inv = 1.0f / fmaxf(sqrtf(ss), 1e-12f);
  float4 o4 = {v.x * inv, v.y * inv, v.z * inv, v.w * inv};
  int l = m >> 5, b = m & 31;
  ((float4*)(out + ((size_t)b * LQ + l) * KD))[lane] = o4;
}

// ------ scores (2x2 blocked): sc[b][q][s] = TEMP * Q[b][q]·K[b][s], stride 672 ------
__global__ void k_scores(const float* __restrict__ Q, const float* __restrict__ Kt,
                         float* __restrict__ sc) {
  int lane = threadIdx.x;
  int half = lane >> 4, l15 = lane & 15;
  int m0 = blockIdx.x * 32, n0 = blockIdx.y * 32, b = blockIdx.z;
  int q0 = m0 + l15;      q0 = q0 < LQ - 1 ? q0 : LQ - 1;
  int q1 = m0 + 16 + l15; q1 = q1 < LQ - 1 ? q1 : LQ - 1;
  int s0 = n0 + l15;      s0 = s0 < LQ - 1 ? s0 : LQ - 1;
  int s1 = n0 + 16 + l15; s1 = s1 < LQ - 1 ? s1 : LQ - 1;
  const float* a0 = Q + ((size_t)b * LQ + q0) * KD + 2 * half;
  const float* a1 = Q + ((size_t)b * LQ + q1) * KD + 2 * half;
  const float* b0 = Kt + ((size_t)b * LQ + s0) * KD + 2 * half;
  const float* b1 = Kt + ((size_t)b * LQ + s1) * KD + 2 * half;
  v8f c00 = {}, c01 = {}, c10 = {}, c11 = {};
  for (int k0 = 0; k0 < KD; k0 += 4) {
    v2f A0 = {a0[k0], a0[k0 + 1]};
    v2f A1 = {a1[k0], a1[k0 + 1]};
    v2f B0 = {b0[k0], b0[k0 + 1]};
    v2f B1 = {b1[k0], b1[k0 + 1]};
    c00 = wmma4(A0, B0, c00);
    c01 = wmma4(A0, B1, c01);
    c10 = wmma4(A1, B0, c10);
    c11 = wmma4(A1, B1, c11);
  }
  float* out = sc + (size_t)b * LPAD * LPAD;
  int mrow = m0 + 8 * half;
  for (int r = 0; r < 8; ++r) {
    out[(size_t)(mrow + r) * LPAD + n0 + l15]           = c00[r] * TEMP;
    out[(size_t)(mrow + r) * LPAD + n0 + 16 + l15]      = c01[r] * TEMP;
    out[(size_t)(mrow + 16 + r) * LPAD + n0 + l15]      = c10[r] * TEMP;
    out[(size_t)(mrow + 16 + r) * LPAD + n0 + 16 + l15] = c11[r] * TEMP;
  }
}

// row softmax over s=0..647 of sc[b][q][*]
__global__ void k_softmax(float* __restrict__ sc) {
  int q = blockIdx.x, b = blockIdx.y, tid = threadIdx.x;
  float* row = sc + (size_t)b * LPAD * LPAD + (size_t)q * LPAD;
  __shared__ float red[256];
  float m = -1e30f;
  for (int s = tid; s < LQ; s += 256) m = fmaxf(m, row[s]);
  red[tid] = m; __syncthreads();
  for (int o = 128; o > 0; o >>= 1) { if (tid < o) red[tid] = fmaxf(red[tid], red[tid + o]); __syncthreads(); }
  m = red[0]; __syncthreads();
  float sum = 0.0f;
  for (int s = tid; s < LQ; s += 256) { float e = expf(row[s] - m); row[s] = e; sum += e; }
  red[tid] = sum; __syncthreads();
  for (int o = 128; o > 0; o >>= 1) { if (tid < o) red[tid] += red[tid + o]; __syncthreads(); }
  float inv = 1.0f / red[0];
  for (int s = tid; s < LQ; s += 256) row[s] *= inv;
}

// mask[q*32+b] = sum_s aff[b][q][s] * p[s*32+b]   (p broadcast over D collapses attn to a scalar)
__global__ void k_mask(const float* __restrict__ sc, const float* __restrict__ p,
                       float* __restrict__ mask) {
  int q = blockIdx.x, b = blockIdx.y, lane = threadIdx.x;
  const float* row = sc + (size_t)b * LPAD * LPAD + (size_t)q * LPAD;
  float s = 0.0f;
  for (int j = lane; j < LQ; j += 32) s += row[j] * p[(size_t)j * 32 + b];
  for (int o = 16; o > 0; o >>= 1) s += __shfl_xor(s, o, 32);
  if (lane == 0) mask[(size_t)q * 32 + b] = s;
}

// ------ PV (2x2 blocked): O[q][b][d] = sum_s aff[b][q][s] * V[s][b][d] ------
__global__ void k_pv(const float* __restrict__ sc, const float* __restrict__ V,
                     float* __restrict__ O) {
  int lane = threadIdx.x;
  int half = lane >> 4, l15 = lane & 15;
  int m0 = blockIdx.x * 32, n0 = blockIdx.y * 32, b = blockIdx.z;
  const float* a0 = sc + (size_t)b * LPAD * LPAD + (size_t)(m0 + l15) * LPAD + 2 * half;
  const float* a1 = sc + (size_t)b * LPAD * LPAD + (size_t)(m0 + 16 + l15) * LPAD + 2 * half;
  const float* v0 = V + (size_t)b * DIM + n0 + l15;        // V[s][b][d] stride BATCH*DIM
  const float* v1 = V + (size_t)b * DIM + n0 + 16 + l15;
  v8f c00 = {}, c01 = {}, c10 = {}, c11 = {};
  for (int k0 = 0; k0 < LQ; k0 += 4) {
    size_t s = (size_t)(k0 + 2 * half);
    if (k0 + 8 < LQ)
      __builtin_prefetch(v0 + (s + 8) * (BATCH * DIM), 0, 0);
    v2f A0 = {a0[k0], a0[k0 + 1]};
    v2f A1 = {a1[k0], a1[k0 + 1]};
    v2f B0 = {v0[s * (BATCH * DIM)], v0[(s + 1) * (BATCH * DIM)]};
    v2f B1 = {v1[s * (BATCH * DIM)], v1[(s + 1) * (BATCH * DIM)]};
    c00 = wmma4(A0, B0, c00);
    c01 = wmma4(A0, B1, c01);
    c10 = wmma4(A1, B0, c10);
    c11 = wmma4(A1, B1, c11);
  }
  int mrow = m0 + 8 * half;
  for (int r = 0; r < 8; ++r) {
    int qa = mrow + r, qb = mrow + 16 + r;
    if (qa < LQ) {
      O[((size_t)qa * BATCH + b) * DIM + n0 + l15]      = c00[r];
      O[((size_t)qa * BATCH + b) * DIM + n0 + 16 + l15] = c01[r];
    }
    if (qb < LQ) {
      O[((size_t)qb * BATCH + b) * DIM + n0 + l15]      = c10[r];
      O[((size_t)qb * BATCH + b) * DIM + n0 + 16 + l15] = c11[r];
    }
  }
}

// ---------------- instance-norm reductions (deterministic block trees) ----------------

// ss[ib*4] = sum over (hw,d) of y^2 for (img i, batch b), ib = i*32+b
__global__ void k_ssq(const float* __restrict__ y, float* __restrict__ ss) {
  int ib = blockIdx.x, tid = threadIdx.x;
  int i = ib >> 5, b = ib & 31;
  __shared__ float red[256];
  float acc = 0.0f;
  for (int j = tid; j < HWPX * DIM; j += 256) {
    int hw = j >> 9, d = j & 511;
    float v = y[((size_t)(i * HWPX + hw) * 32 + b) * 512 + d];
    acc += v * v;
  }
  red[tid] = acc; __syncthreads();
  for (int o = 128; o > 0; o >>= 1) { if (tid < o) red[tid] += red[tid + o]; __syncthreads(); }
  if (tid == 0) ss[ib * 4] = red[0];
}

// x = y * NORM_SCALE*sqrt(N/(ss+eps))
__global__ void k_scaleinto(const float* __restrict__ y, const float* __restrict__ ss,
                            float* __restrict__ x) {
  size_t o = (size_t)blockIdx.x * blockDim.x + threadIdx.x;
  if (o >= NELEM) return;
  int l = (int)(o >> 14);
  int ib = (l / HWPX) * 32 + (int)((o >> 9) & 31);
  float s = NORM_SCALE * sqrtf(INORM_N / (ss[ib * 4] + 1e-5f));
  x[o] = y[o] * s;
}

// three reductions for the fused triple-inorm:
// A=||x*mask||^2, C=||x+t3||^2, M=<x*mask, x+t3> per (img,batch)
__global__ void k_comb_red(const float* __restrict__ x, const float* __restrict__ t3,
                           const float* __restrict__ mask, float* __restrict__ ss) {
  int ib = blockIdx.x, tid = threadIdx.x;
  int i = ib >> 5, b = ib & 31;
  __shared__ float red[256];
  float aA = 0.0f, aC = 0.0f, aM = 0.0f;
  for (int j = tid; j < HWPX * DIM; j += 256) {
    int hw = j >> 9, d = j & 511;
    int l = i * HWPX + hw;
    size_t o = ((size_t)l * 32 + b) * 512 + d;
    float xm = x[o] * mask[(size_t)l * 32 + b];
    float xt = x[o] + t3[o];
    aA += xm * xm; aC += xt * xt; aM += xm * xt;
  }
  red[tid] = aA; __syncthreads();
  for (int o = 128; o > 0; o >>= 1) { if (tid < o) red[tid] += red[tid + o]; __syncthreads(); }
  if (tid == 0) ss[ib * 4 + 0] = red[0];
  __syncthreads();
  red[tid] = aC; __syncthreads();
  for (int o = 128; o > 0; o >>= 1) { if (tid < o) red[tid] += red[tid + o]; __syncthreads(); }
  if (tid == 0) ss[ib * 4 + 1] = red[0];
  __syncthreads();
  red[tid] = aM; __syncthreads();
  for (int o = 128; o > 0; o >>= 1) { if (tid < o) red[tid] += red[tid + o]; __syncthreads(); }
  if (tid == 0) ss[ib * 4 + 2] = red[0];
}

// x = ( s2*(x*mask) + s4*(x+t3) ) * sf ;  sf from ||s2*xm+s4*xt||^2 expanded analytically
__global__ void k_comb_scale(float* __restrict__ x, const float* __restrict__ t3,
                             const float* __restrict__ mask, const float* __restrict__ ss) {
  size_t o = (size_t)blockIdx.x * blockDim.x + threadIdx.x;
  if (o >= NELEM) return;
  int l = (int)(o >> 14);
  int ib = (l / HWPX) * 32 + (int)((o >> 9) & 31);
  float A = ss[ib * 4 + 0], C = ss[ib * 4 + 1], M = ss[ib * 4 + 2];
  float s2 = NORM_SCALE * sqrtf(INORM_N / (A + 1e-5f));
  float s4 = NORM_SCALE * sqrtf(INORM_N / (C + 1e-5f));
  float ssy = s2 * s2 * A + 2.0f * s2 * s4 * M + s4 * s4 * C;
  float sf = NORM_SCALE * sqrtf(INORM_N / (ssy + 1e-5f));
  float xm = x[o] * mask[o >> 9];
  float xt = x[o] + t3[o];
  x[o] = (s2 * xm + s4 * xt) * sf;
}

// out = [ x flat (L,B,D) | out_feat (n*b, d, h, w) ]
__global__ void k_out(const float* __restrict__ x, float* __restrict__ out) {
  size_t o = (size_t)blockIdx.x * blockDim.x + threadIdx.x;
  if (o >= NELEM) return;
  out[o] = x[o];
  int d = (int)(o & 511);
  int b = (int)((o >> 9) & 31);
  int l = (int)(o >> 14);
  int i = l / HWPX, hw = l - i * HWPX;
  out[(size_t)NELEM + ((size_t)(i * 32 + b) * 512 + d) * HWPX + hw] = x[o];
}

// ---------------- launcher ----------------
extern "C" void kernel_launch(void* const* d_in, const int* in_sizes, int n_in,
                              void* d_out, int out_size, void* d_ws, size_t ws_size,
                              hipStream_t stream) {
  const float* tgt  = (const float*)d_in[0];
  const float* pe   = (const float*)d_in[1];
  const float* mem  = (const float*)d_in[2];
  const float* pos  = (const float*)d_in[3];
  const float* Wks  = (const float*)d_in[4];
  const float* bks  = (const float*)d_in[5];
  const float* Wkc  = (const float*)d_in[6];
  const float* bkc  = (const float*)d_in[7];
  float* out = (float*)d_out;
  float* ws  = (float*)d_ws;

  const size_t NQ = (size_t)MROWS * KD;           // 2,654,208
  float* x    = ws;                               // [L][B][D]
  float* y    = x + NELEM;                        // attn out / temp
  float* t3   = y + NELEM;
  float* memp = t3 + NELEM;
  float* qs   = memp + NELEM;                     // [B][L][K]
  float* qc   = qs + NQ;
  float* wkm  = qc + NQ;
  float* raw  = wkm + NQ;                         // [M][K]
  float* sc   = raw + NQ;                         // [B][672][672]
  float* p    = sc + (size_t)BATCH * LPAD * LPAD; // [L][B]
  float* mask = p + LQ * BATCH;
  float* ssb  = mask + LQ * BATCH;                // 64*4 reduction slots
  // total ~272 MB of workspace

  const int EW = (NELEM + 255) / 256;
  dim3 gProj(MROWS / 32, KD / 32);                // 648 x 4, 32x32 per wave
  dim3 gSc(21, 21, BATCH);                        // covers 672x672 (clamped loads)
  dim3 gPV(21, DIM / 32, BATCH);
  dim3 gRow(LQ, BATCH);

  // one-time setup
  k_flatten<<<EW, 256, 0, stream>>>(tgt, x);
  k_pos<<<(LQ * BATCH + 255) / 256, 256, 0, stream>>>(pos, p);
  k_memp<<<EW, 256, 0, stream>>>(mem, p, memp);
  k_proj<<<gProj, 32, 0, stream>>>(mem, Wkc, bkc, raw);     // wk_cross(memory): layer-invariant
  k_l2row<<<MROWS / 4, 128, 0, stream>>>(raw, wkm);

  for (int layer = 0; layer < 6; ++layer) {
    k_addpe<<<EW, 256, 0, stream>>>(x, pe);
    // self attention (q = k = v = x)
    k_proj<<<gProj, 32, 0, stream>>>(x, Wks, bks, raw);
    k_l2row<<<MROWS / 4, 128, 0, stream>>>(raw, qs);
    k_scores<<<gSc, 32, 0, stream>>>(qs, qs, sc);
    k_softmax<<<gRow, 256, 0, stream>>>(sc);
    k_pv<<<gPV, 32, 0, stream>>>(sc, x, y);
    k_addinto<<<EW, 256, 0, stream>>>(y, x);
    k_ssq<<<64, 256, 0, stream>>>(y, ssb);
    k_scaleinto<<<EW, 256, 0, stream>>>(y, ssb, x);
    // cross attention (shared affinity for mask + t3)
    k_proj<<<gProj, 32, 0, stream>>>(x, Wkc, bkc, raw);
    k_l2row<<<MROWS / 4, 128, 0, stream>>>(raw, qc);
    k_scores<<<gSc, 32, 0, stream>>>(qc, wkm, sc);
    k_softmax<<<gRow, 256, 0, stream>>>(sc);
    k_mask<<<gRow, 32, 0, stream>>>(sc, p, mask);
    k_pv<<<gPV, 32, 0, stream>>>(sc, memp, t3);
    // fused t2/t4/final inorm
    k_comb_red<<<64, 256, 0, stream>>>(x, t3, mask, ssb);
    k_comb_scale<<<EW, 256, 0, stream>>>(x, t3, mask, ssb);
  }
  k_out<<<EW, 256, 0, stream>>>(x, out);
}